// Attention_214748364819
// MI455X (gfx1250) — compile-verified
//
#include <hip/hip_runtime.h>
#include <math.h>

// ---------------------------------------------------------------------------
// Attention: h = dec@W.T + b ; scores = einsum('bsd,bd->bs'); softmax; context
// B=64, S=4096, Dd=1024, De=512, fp32.
// Strategy: WMMA f32 GEMM for h; fused single-pass online-softmax over the
// 512 MiB encoder tensor (read exactly once, nontemporal); small reduce.
// ---------------------------------------------------------------------------

typedef float v2f __attribute__((ext_vector_type(2)));
typedef float v4f __attribute__((ext_vector_type(4)));
typedef float v8f __attribute__((ext_vector_type(8)));

#define BB   64
#define SS   4096
#define DD   1024
#define DE   512
#define PSTRIDE 520   // 512 acc + m + l + pad (floats)

// ---------------------------------------------------------------------------
// Kernel 1: h[64,512] = dec[64,1024] @ W.T[1024,512] + bias, via
// V_WMMA_F32_16X16X4_F32. One 16x16 output tile per wave; 8 waves per block.
// A 16x4 layout: lanes 0-15 -> M=lane, V0=K0,V1=K1 ; lanes 16-31 -> K2,K3.
// B 4x16 layout: lanes 0-15 -> K0(V0),K1(V1) ; lanes 16-31 -> K2(V0),K3(V1),
//                N = lane & 15. B[k,n] = W[n,k] (row-major W gives contiguous
//                k-pairs per lane).
// ---------------------------------------------------------------------------
__global__ __launch_bounds__(256) void linear_wmma(
    const float* __restrict__ dec, const float* __restrict__ W,
    const float* __restrict__ bias, float* __restrict__ h) {
  const int wave = threadIdx.x >> 5;
  const int lane = threadIdx.x & 31;
  const int half = lane >> 4;
  const int l16  = lane & 15;

  const int tile = blockIdx.x * 8 + wave;   // 0..127
  const int mt = tile >> 5;                 // 0..3   (M tiles of 16)
  const int nt = tile & 31;                 // 0..31  (N tiles of 16)

  const int mrow = mt * 16 + l16;           // A row for this lane
  const int ncol = nt * 16 + l16;           // B col / D col for this lane

  const float* arow = dec + (size_t)mrow * DD + 2 * half;
  const float* brow = W   + (size_t)ncol * DD + 2 * half;

  v8f c = {};
  #pragma unroll 4
  for (int k0 = 0; k0 < DD; k0 += 4) {
    v2f a  = *(const v2f*)(arow + k0);
    v2f bb = *(const v2f*)(brow + k0);
    c = __builtin_amdgcn_wmma_f32_16x16x4_f32(
        /*neg_a=*/false, a, /*neg_b=*/false, bb,
        /*c_mod=*/(short)0, c, /*reuse_a=*/false, /*reuse_b=*/false);
  }

  const float bv = bias[ncol];
  // D layout: VGPR r -> lanes 0-15: M = r, lanes 16-31: M = r + 8 ; N = l16
  #pragma unroll
  for (int r = 0; r < 8; ++r) {
    const int m = mt * 16 + r + 8 * half;
    h[(size_t)m * DE + ncol] = c[r] + bv;
  }
}

// ---------------------------------------------------------------------------
// Kernel 2: fused scores + online softmax + weighted accumulation.
// Grid: (C chunks, 64 batches). Block: 256 threads = 8 waves.
// Each wave processes every 8th s in the chunk; lane owns d = 128*j + 4*lane
// (+0..3), j=0..3 -> 16 of the 512 features, fully coalesced v4f loads.
// Online softmax state (m, l, acc[16]) per wave; block-combined via LDS.
// ---------------------------------------------------------------------------
__global__ __launch_bounds__(256) void attn_partial(
    const float* __restrict__ enc, const float* __restrict__ h,
    float* __restrict__ partials, int C) {
  const int b     = blockIdx.y;
  const int chunk = blockIdx.x;
  const int Sc    = SS / C;
  const int s0    = chunk * Sc;
  const int wave  = threadIdx.x >> 5;
  const int lane  = threadIdx.x & 31;

  // h[b, d-slots for this lane]
  v4f hreg[4];
  const float* hb = h + (size_t)b * DE;
  #pragma unroll
  for (int j = 0; j < 4; ++j)
    hreg[j] = *(const v4f*)(hb + j * 128 + lane * 4);

  v4f acc[4];
  #pragma unroll
  for (int j = 0; j < 4; ++j) acc[j] = (v4f){0.f, 0.f, 0.f, 0.f};
  float m = -__builtin_inff();
  float l = 0.f;

  for (int s = s0 + wave; s < s0 + Sc; s += 8) {
    const float* row = enc + ((size_t)s * BB + b) * DE;
    v4f e[4];
    #pragma unroll
    for (int j = 0; j < 4; ++j)
      e[j] = __builtin_nontemporal_load((const v4f*)(row + j * 128 + lane * 4));

    float dot = 0.f;
    #pragma unroll
    for (int j = 0; j < 4; ++j) {
      dot = fmaf(e[j].x, hreg[j].x, dot);
      dot = fmaf(e[j].y, hreg[j].y, dot);
      dot = fmaf(e[j].z, hreg[j].z, dot);
      dot = fmaf(e[j].w, hreg[j].w, dot);
    }
    // wave32 butterfly reduce -> score uniform on all lanes
    #pragma unroll
    for (int off = 16; off >= 1; off >>= 1)
      dot += __shfl_xor(dot, off, 32);
    const float score = dot;

    if (score > m) {               // wave-uniform branch (rare after warmup)
      const float sc = __expf(m - score);
      l = fmaf(l, sc, 1.0f);
      #pragma unroll
      for (int j = 0; j < 4; ++j) {
        acc[j].x = fmaf(acc[j].x, sc, e[j].x);
        acc[j].y = fmaf(acc[j].y, sc, e[j].y);
        acc[j].z = fmaf(acc[j].z, sc, e[j].z);
        acc[j].w = fmaf(acc[j].w, sc, e[j].w);
      }
      m = score;
    } else {                       // common path: 1 exp + 16 FMA per row
      const float p = __expf(score - m);
      l += p;
      #pragma unroll
      for (int j = 0; j < 4; ++j) {
        acc[j].x = fmaf(p, e[j].x, acc[j].x);
        acc[j].y = fmaf(p, e[j].y, acc[j].y);
        acc[j].z = fmaf(p, e[j].z, acc[j].z);
        acc[j].w = fmaf(p, e[j].w, acc[j].w);
      }
    }
  }

  // ---- deterministic block combine across the 8 waves ----
  __shared__ float s_m[8];
  __shared__ float s_l[8];
  __shared__ float s_acc[8 * DE];

  if (lane == 0) { s_m[wave] = m; s_l[wave] = l; }
  __syncthreads();

  float M = s_m[0];
  #pragma unroll
  for (int w = 1; w < 8; ++w) M = fmaxf(M, s_m[w]);

  const float myscale = __expf(m - M);
  #pragma unroll
  for (int j = 0; j < 4; ++j) {
    v4f v = acc[j];
    v.x *= myscale; v.y *= myscale; v.z *= myscale; v.w *= myscale;
    *(v4f*)(&s_acc[wave * DE + j * 128 + lane * 4]) = v;
  }
  __syncthreads();

  float* out = partials + ((size_t)b * C + chunk) * PSTRIDE;
  const int tid = threadIdx.x;
  #pragma unroll
  for (int d = tid; d < DE; d += 256) {
    float sum = 0.f;
    #pragma unroll
    for (int w = 0; w < 8; ++w) sum += s_acc[w * DE + d];
    out[d] = sum;
  }
  if (tid == 0) {
    float L = 0.f;
    #pragma unroll
    for (int w = 0; w < 8; ++w) L = fmaf(s_l[w], __expf(s_m[w] - M), L);
    out[DE]     = M;
    out[DE + 1] = L;
  }
}

// ---------------------------------------------------------------------------
// Kernel 3: merge chunk partials per batch; write context[64,512].
// ---------------------------------------------------------------------------
__global__ __launch_bounds__(256) void attn_reduce(
    const float* __restrict__ partials, float* __restrict__ ctx, int C) {
  const int b = blockIdx.x;
  const int tid = threadIdx.x;
  const float* pb = partials + (size_t)b * C * PSTRIDE;

  float M = -__builtin_inff();
  for (int c = 0; c < C; ++c) M = fmaxf(M, pb[(size_t)c * PSTRIDE + DE]);

  __shared__ float s_sc[64];
  if (tid < C) s_sc[tid] = __expf(pb[(size_t)tid * PSTRIDE + DE] - M);
  __syncthreads();

  float L = 0.f;
  for (int c = 0; c < C; ++c) L = fmaf(pb[(size_t)c * PSTRIDE + DE + 1], s_sc[c], L);
  const float inv = 1.0f / L;

  for (int d = tid; d < DE; d += 256) {
    float sum = 0.f;
    for (int c = 0; c < C; ++c)
      sum = fmaf(pb[(size_t)c * PSTRIDE + d], s_sc[c], sum);
    ctx[(size_t)b * DE + d] = sum * inv;
  }
}

// ---------------------------------------------------------------------------
// Host launcher
// inputs: 0=decoder_hidden[64,1024] f32, 1=encoder_output[4096,64,512] f32,
//         2=W[512,1024] f32, 3=b[512] f32 ; output: context[64,512] f32
// ---------------------------------------------------------------------------
extern "C" void kernel_launch(void* const* d_in, const int* in_sizes, int n_in,
                              void* d_out, int out_size, void* d_ws, size_t ws_size,
                              hipStream_t stream) {
  const float* dec  = (const float*)d_in[0];
  const float* enc  = (const float*)d_in[1];
  const float* W    = (const float*)d_in[2];
  const float* bias = (const float*)d_in[3];
  float* ctx = (float*)d_out;

  // workspace: h[64*512] floats, then partials[64 * C * PSTRIDE] floats
  const size_t h_bytes = (size_t)BB * DE * sizeof(float);
  int C = 64;  // chunks along S; pick largest that fits the workspace
  while (C > 1 && h_bytes + (size_t)BB * C * PSTRIDE * sizeof(float) > ws_size)
    C >>= 1;

  float* h        = (float*)d_ws;
  float* partials = (float*)((char*)d_ws + h_bytes);

  linear_wmma<<<16, 256, 0, stream>>>(dec, W, bias, h);
  attn_partial<<<dim3(C, BB), 256, 0, stream>>>(enc, h, partials, C);
  attn_reduce<<<BB, 256, 0, stream>>>(partials, ctx, C);
}